// Adj_Generator_48043504173314
// MI455X (gfx1250) — compile-verified
//
#include <hip/hip_runtime.h>

// Problem constants from the reference: B=4096, V=64, F=128, HO=3
#define V_DIM 64
#define F_DIM 128
#define TILE  (V_DIM * F_DIM)   // 8192 floats = 32 KB per tensor per batch
#define NT    256               // 8 wave32 per workgroup
#define CHUNKS (TILE / 4 / NT)  // 8 float4 chunks per thread

typedef float v4f __attribute__((ext_vector_type(4)));
typedef int   v4i __attribute__((ext_vector_type(4)));

// Address-space-qualified pointer types for the async builtins:
// diagnostic showed the params are int4* in AS1 (global) / AS3 (LDS).
typedef __attribute__((address_space(1))) void  gvoid;
typedef __attribute__((address_space(3))) void  lvoid;
typedef __attribute__((address_space(1))) v4i   gv4i;
typedef __attribute__((address_space(3))) v4i   lv4i;

#if defined(__has_builtin)
#  if __has_builtin(__builtin_amdgcn_global_load_async_to_lds_b128)
#    define HAVE_ASYNC_LD 1
#  endif
#  if __has_builtin(__builtin_amdgcn_global_store_async_from_lds_b128)
#    define HAVE_ASYNC_ST 1
#  endif
#  if __has_builtin(__builtin_amdgcn_s_wait_asynccnt)
#    define WAIT_ASYNC() __builtin_amdgcn_s_wait_asynccnt(0)
#  endif
#endif
#ifndef WAIT_ASYNC
#  define WAIT_ASYNC() asm volatile("s_wait_asynccnt 0x0" ::: "memory")
#endif

__global__ __launch_bounds__(NT) void adj_generator_kernel(
    const float* __restrict__ sm,      // softmax   [B, V, F]
    const float* __restrict__ lp,      // log_probs [B, V, F]
    float* __restrict__ out_sm,        // pass-through softmax [B, V, F]
    float* __restrict__ out_adj,       // cond_adj as 0.0/1.0  [B, V, F]
    float* __restrict__ out_ent)       // entropy              [B]
{
    __shared__ float s_s[TILE];        // staged softmax tile (32 KB)
    __shared__ float s_wsum[NT / 32];  // per-wave entropy partials
    __shared__ int   s_i0[F_DIM], s_i1[F_DIM], s_i2[F_DIM];

    const int b   = blockIdx.x;
    const int tid = threadIdx.x;

    const float* sm_b  = sm  + (size_t)b * TILE;
    const float* lp_b  = lp  + (size_t)b * TILE;
    float* osm_b  = out_sm  + (size_t)b * TILE;
    float* oadj_b = out_adj + (size_t)b * TILE;

    // ---------------- Phase 1: DMA softmax tile HBM -> LDS (ASYNCcnt path) ----
#if defined(HAVE_ASYNC_LD)
    {
        // Each lane moves 16B per issue; INST_OFFSET advances both addresses.
        gv4i* gsrc = (gv4i*)(gvoid*)(void*)(sm_b + tid * 4);
        lv4i* ldst = (lv4i*)(lvoid*)(void*)(&s_s[tid * 4]);
#define ALD(off) __builtin_amdgcn_global_load_async_to_lds_b128(gsrc, ldst, (off), 0)
        ALD(0);     ALD(4096);  ALD(8192);  ALD(12288);
        ALD(16384); ALD(20480); ALD(24576); ALD(28672);
#undef ALD
    }
#else
    #pragma unroll
    for (int k = 0; k < CHUNKS; ++k) {
        int c = tid + k * NT;
        ((v4f*)s_s)[c] = ((const v4f*)sm_b)[c];
    }
#endif

    // Overlap the DMA with streaming log_probs into registers (read-once -> NT).
    v4f lv[CHUNKS];
    #pragma unroll
    for (int k = 0; k < CHUNKS; ++k) {
        int c = tid + k * NT;
        lv[k] = __builtin_nontemporal_load(((const v4f*)lp_b) + c);
    }

#if defined(HAVE_ASYNC_LD)
    WAIT_ASYNC();
#endif
    __syncthreads();   // all waves' tile portions visible in LDS

    // ---------------- Phase 2: pass-through copy + entropy ------------------
#if defined(HAVE_ASYNC_ST)
    {
        // softmax output is an exact copy of the staged tile: DMA LDS -> HBM,
        // no VGPR round trip.
        gv4i* gdst = (gv4i*)(gvoid*)(void*)(osm_b + tid * 4);
        lv4i* lsrc = (lv4i*)(lvoid*)(void*)(&s_s[tid * 4]);
#define AST(off) __builtin_amdgcn_global_store_async_from_lds_b128(gdst, lsrc, (off), 0)
        AST(0);     AST(4096);  AST(8192);  AST(12288);
        AST(16384); AST(20480); AST(24576); AST(28672);
#undef AST
    }
#endif

    v4f sv[CHUNKS];
    float ep = 0.0f;
    #pragma unroll
    for (int k = 0; k < CHUNKS; ++k) {
        int c = tid + k * NT;
        v4f s4 = ((const v4f*)s_s)[c];
        v4f l4 = lv[k];
        sv[k] = s4;
        ep -= s4.x * l4.x;
        ep -= s4.y * l4.y;
        ep -= s4.z * l4.z;
        ep -= s4.w * l4.w;
#if !defined(HAVE_ASYNC_ST)
        __builtin_nontemporal_store(s4, ((v4f*)osm_b) + c);
#endif
    }

    // wave32 reduction of entropy partials
    #pragma unroll
    for (int off = 16; off > 0; off >>= 1)
        ep += __shfl_down(ep, off, 32);
    if ((tid & 31) == 0) s_wsum[tid >> 5] = ep;
    __syncthreads();

    // ---------------- Phase 3: per-factor top-3 over V (column scan in LDS) --
    if (tid < F_DIM) {
        const int f = tid;
        float v0 = -1e30f, v1 = -1e30f, v2 = -1e30f;
        int   i0 = 0, i1 = 0, i2 = 0;
        #pragma unroll 4
        for (int v = 0; v < V_DIM; ++v) {
            float x = s_s[v * F_DIM + f];   // lane f -> bank f%64: conflict-free
            if (x > v0)      { v2 = v1; i2 = i1; v1 = v0; i1 = i0; v0 = x; i0 = v; }
            else if (x > v1) { v2 = v1; i2 = i1; v1 = x;  i1 = v; }
            else if (x > v2) { v2 = x;  i2 = v; }
        }
        float p3 = v0 * v0 * v0;
        float p2 = 3.0f * v1 * v2 * (v1 + v2);
        float p1 = 6.0f * v0 * v1 * v2;
        bool c3 = (p3 > p2) && (p3 > p1);
        bool c2 = (p2 >= p3) && (p2 > p1);
        s_i0[f] = i0;
        s_i1[f] = c3 ? i0 : i1;
        s_i2[f] = (c3 || c2) ? i0 : i2;
    }
    if (tid == 0) {
        float t = 0.0f;
        #pragma unroll
        for (int w = 0; w < NT / 32; ++w) t += s_wsum[w];
        out_ent[b] = t * (1.0f / (float)F_DIM);  // sum over V, mean over F
    }
    __syncthreads();

    // ---------------- Phase 4: cond_adj = (s > 0.01) & membership ------------
    #pragma unroll
    for (int k = 0; k < CHUNKS; ++k) {
        int c  = tid + k * NT;
        int n  = c * 4;
        int v  = n >> 7;        // row (variable index)
        int f0 = n & (F_DIM - 1);
        v4f s4 = sv[k];
        v4f o;
        o.x = (s4.x > 0.01f && (v == s_i0[f0+0] || v == s_i1[f0+0] || v == s_i2[f0+0])) ? 1.0f : 0.0f;
        o.y = (s4.y > 0.01f && (v == s_i0[f0+1] || v == s_i1[f0+1] || v == s_i2[f0+1])) ? 1.0f : 0.0f;
        o.z = (s4.z > 0.01f && (v == s_i0[f0+2] || v == s_i1[f0+2] || v == s_i2[f0+2])) ? 1.0f : 0.0f;
        o.w = (s4.w > 0.01f && (v == s_i0[f0+3] || v == s_i1[f0+3] || v == s_i2[f0+3])) ? 1.0f : 0.0f;
        __builtin_nontemporal_store(o, ((v4f*)oadj_b) + c);
    }
}

extern "C" void kernel_launch(void* const* d_in, const int* in_sizes, int n_in,
                              void* d_out, int out_size, void* d_ws, size_t ws_size,
                              hipStream_t stream) {
    const float* sm = (const float*)d_in[0];   // softmax   [B, V, F]
    const float* lp = (const float*)d_in[1];   // log_probs [B, V, F]
    float* out = (float*)d_out;

    const int nb = in_sizes[0] / TILE;         // B
    const size_t N = (size_t)nb * TILE;        // B*V*F

    // d_out layout (reference return order, flat): softmax | cond_adj | ent
    adj_generator_kernel<<<nb, NT, 0, stream>>>(sm, lp, out, out + N, out + 2 * N);
}